// GCN_34316788695414
// MI455X (gfx1250) — compile-verified
//
#include <hip/hip_runtime.h>
#include <cstdint>

typedef float v2f __attribute__((ext_vector_type(2)));
typedef float v8f __attribute__((ext_vector_type(8)));
typedef long long i64;

// ---------------------------------------------------------------------------
// Degree / normalization
// ---------------------------------------------------------------------------
__global__ void fill_ones(float* __restrict__ p, int n) {
    int i = blockIdx.x * blockDim.x + threadIdx.x;
    if (i < n) p[i] = 1.0f;                       // self-loop contributes 1
}

__global__ void degree_edges(const i64* __restrict__ dst, float* __restrict__ deg, int e) {
    int i = blockIdx.x * blockDim.x + threadIdx.x;
    if (i < e) atomicAdd(&deg[(int)dst[i]], 1.0f);
}

__global__ void rsqrt_inplace(float* __restrict__ p, int n) {
    int i = blockIdx.x * blockDim.x + threadIdx.x;
    if (i < n) p[i] = rsqrtf(p[i]);               // deg >= 1 always (self-loop)
}

// ---------------------------------------------------------------------------
// WMMA GEMM: H[N x OUTC] = (RELU? relu(A) : A)[N x 128] @ W[128 x OUTC]
// Block = 64 threads (2 waves); each wave computes one 16x16 tile.
// Uses V_WMMA_F32_16X16X4_F32 (full fp32 precision, CDNA5 matrix pipe).
// A 16x4 f32 fragment layout (ISA 7.12.2): lanes 0-15 hold K=0,1 in v[0],v[1];
// lanes 16-31 hold K=2,3. B mirrored. C/D: VGPR v -> row v (+8 for hi lanes).
// ---------------------------------------------------------------------------
template <int OUTC, bool RELU>
__global__ void __launch_bounds__(64)
gemm_wmma(const float* __restrict__ A, const float* __restrict__ W,
          float* __restrict__ H) {
    __shared__ float lw[128 * 16];                // W[:, col0:col0+16]
    const int col0 = blockIdx.y * 16;
    for (int idx = threadIdx.x; idx < 128 * 16; idx += 64) {
        int k = idx >> 4, n = idx & 15;
        lw[idx] = W[k * OUTC + col0 + n];
    }
    __syncthreads();

    const int wave = threadIdx.x >> 5;
    const int lane = threadIdx.x & 31;
    const int row0 = blockIdx.x * 32 + wave * 16;
    const int m    = lane & 15;
    const int kh   = (lane >> 4) << 1;            // 0 (lo lanes) or 2 (hi lanes)
    const float* arow = A + (size_t)(row0 + m) * 128;

    v8f c = {};
    #pragma unroll
    for (int k0 = 0; k0 < 128; k0 += 4) {
        v2f a = *(const v2f*)(arow + k0 + kh);    // K = k0+kh, k0+kh+1
        if (RELU) { a.x = fmaxf(a.x, 0.0f); a.y = fmaxf(a.y, 0.0f); }
        v2f b;
        b.x = lw[(k0 + kh) * 16 + m];
        b.y = lw[(k0 + kh + 1) * 16 + m];
        c = __builtin_amdgcn_wmma_f32_16x16x4_f32(
                /*neg_a=*/false, a, /*neg_b=*/false, b,
                /*c_mod=*/(short)0, c, /*reuse_a=*/false, /*reuse_b=*/false);
    }

    const int rbase = row0 + ((lane >> 4) << 3);  // +8 rows for hi half-wave
    #pragma unroll
    for (int v = 0; v < 8; ++v)
        H[(size_t)(rbase + v) * OUTC + col0 + m] = c[v];
}

// ---------------------------------------------------------------------------
// out[i,:] = b + h[i,:] * dinv[i]^2   (self-loop term; also zero-free init)
// ---------------------------------------------------------------------------
template <int C>
__global__ void selfloop_bias(const float* __restrict__ h, const float* __restrict__ b,
                              const float* __restrict__ dinv, float* __restrict__ out,
                              int n) {
    int idx = blockIdx.x * blockDim.x + threadIdx.x;
    if (idx >= n * C) return;
    int i = idx / C, cc = idx - i * C;
    float d = dinv[i];
    out[idx] = b[cc] + h[idx] * d * d;
}

// ---------------------------------------------------------------------------
// Edge scatter-add, C=128: one wave per edge; lane covers 4 consecutive
// floats (coalesced 512B b128 gather per wave, f32 atomics to L2).
// ---------------------------------------------------------------------------
__global__ void edge_agg128(const float* __restrict__ h, const float* __restrict__ dinv,
                            const i64* __restrict__ src, const i64* __restrict__ dst,
                            float* __restrict__ out, int e) {
    int wid  = (blockIdx.x * blockDim.x + threadIdx.x) >> 5;
    int lane = threadIdx.x & 31;
    if (wid >= e) return;
    int s = (int)src[wid], d = (int)dst[wid];
    float nrm = dinv[s] * dinv[d];
    float4 v = *(const float4*)(h + (size_t)s * 128 + lane * 4);
    float* op = out + (size_t)d * 128 + lane * 4;
    atomicAdd(op + 0, v.x * nrm);
    atomicAdd(op + 1, v.y * nrm);
    atomicAdd(op + 2, v.z * nrm);
    atomicAdd(op + 3, v.w * nrm);
}

// Edge scatter-add, C=32: one wave per edge, one float per lane.
__global__ void edge_agg32(const float* __restrict__ h, const float* __restrict__ dinv,
                           const i64* __restrict__ src, const i64* __restrict__ dst,
                           float* __restrict__ out, int e) {
    int wid  = (blockIdx.x * blockDim.x + threadIdx.x) >> 5;
    int lane = threadIdx.x & 31;
    if (wid >= e) return;
    int s = (int)src[wid], d = (int)dst[wid];
    float nrm = dinv[s] * dinv[d];
    float v = h[(size_t)s * 32 + lane];
    atomicAdd(out + (size_t)d * 32 + lane, v * nrm);
}

// ---------------------------------------------------------------------------
// log_softmax over 32 classes: one wave per row, one class per lane (wave32).
// ---------------------------------------------------------------------------
__global__ void logsoftmax32(const float* __restrict__ in, float* __restrict__ out, int n) {
    int wid  = (blockIdx.x * blockDim.x + threadIdx.x) >> 5;
    int lane = threadIdx.x & 31;
    if (wid >= n) return;
    float v = in[(size_t)wid * 32 + lane];
    float m = v;
    #pragma unroll
    for (int off = 16; off > 0; off >>= 1) m = fmaxf(m, __shfl_xor(m, off, 32));
    float ex = expf(v - m);
    float s = ex;
    #pragma unroll
    for (int off = 16; off > 0; off >>= 1) s += __shfl_xor(s, off, 32);
    out[(size_t)wid * 32 + lane] = v - m - logf(s);
}

// ---------------------------------------------------------------------------
// Orchestration
// ---------------------------------------------------------------------------
extern "C" void kernel_launch(void* const* d_in, const int* in_sizes, int n_in,
                              void* d_out, int out_size, void* d_ws, size_t ws_size,
                              hipStream_t stream) {
    const float* x  = (const float*)d_in[0];
    const i64*   ei = (const i64*)d_in[1];          // int64 per reference
    const float* W1 = (const float*)d_in[2]; const float* b1 = (const float*)d_in[3];
    const float* W2 = (const float*)d_in[4]; const float* b2 = (const float*)d_in[5];
    const float* W3 = (const float*)d_in[6]; const float* b3 = (const float*)d_in[7];
    const float* W4 = (const float*)d_in[8]; const float* b4 = (const float*)d_in[9];

    const int N = in_sizes[0] / 128;                // 100000
    const int E = in_sizes[1] / 2;                  // 3200000
    const i64* src = ei;
    const i64* dst = ei + E;

    // workspace: dinv[N] | hbuf[N*128] | aggbuf[N*128]
    float* dinv   = (float*)d_ws;
    float* hbuf   = dinv + ((N + 255) & ~255);
    float* aggbuf = hbuf + (size_t)N * 128;

    const int T = 256;
    auto cdiv = [](long long a, long long b) { return (int)((a + b - 1) / b); };

    // degree -> dinv
    fill_ones<<<cdiv(N, T), T, 0, stream>>>(dinv, N);
    degree_edges<<<cdiv(E, T), T, 0, stream>>>(dst, dinv, E);
    rsqrt_inplace<<<cdiv(N, T), T, 0, stream>>>(dinv, N);

    dim3 g128(N / 32, 128 / 16);
    dim3 g32 (N / 32,  32 / 16);
    int edgeBlocks = cdiv((long long)E * 32, T);    // one wave per edge
    int rowWaves   = cdiv((long long)N * 32, T);    // one wave per row

    // layer 1
    gemm_wmma<128, false><<<g128, 64, 0, stream>>>(x, W1, hbuf);
    selfloop_bias<128><<<cdiv((long long)N * 128, T), T, 0, stream>>>(hbuf, b1, dinv, aggbuf, N);
    edge_agg128<<<edgeBlocks, T, 0, stream>>>(hbuf, dinv, src, dst, aggbuf, E);

    // layer 2 (ReLU fused into A-load)
    gemm_wmma<128, true><<<g128, 64, 0, stream>>>(aggbuf, W2, hbuf);
    selfloop_bias<128><<<cdiv((long long)N * 128, T), T, 0, stream>>>(hbuf, b2, dinv, aggbuf, N);
    edge_agg128<<<edgeBlocks, T, 0, stream>>>(hbuf, dinv, src, dst, aggbuf, E);

    // layer 3
    gemm_wmma<128, true><<<g128, 64, 0, stream>>>(aggbuf, W3, hbuf);
    selfloop_bias<128><<<cdiv((long long)N * 128, T), T, 0, stream>>>(hbuf, b3, dinv, aggbuf, N);
    edge_agg128<<<edgeBlocks, T, 0, stream>>>(hbuf, dinv, src, dst, aggbuf, E);

    // layer 4 (OUTC = 32)
    gemm_wmma<32, true><<<g32, 64, 0, stream>>>(aggbuf, W4, hbuf);
    selfloop_bias<32><<<cdiv((long long)N * 32, T), T, 0, stream>>>(hbuf, b4, dinv, aggbuf, N);
    edge_agg32<<<edgeBlocks, T, 0, stream>>>(hbuf, dinv, src, dst, aggbuf, E);

    // log_softmax -> d_out
    logsoftmax32<<<rowWaves, T, 0, stream>>>(aggbuf, (float*)d_out, N);
}